// bSpline_76630806495781
// MI455X (gfx1250) — compile-verified
//
#include <hip/hip_runtime.h>
#include <hip/hip_bf16.h>
#include <stdint.h>

#define BLK 256

// ---- CDNA5 async global->LDS copy (16B per lane), GVS addressing:
//      mem = SGPR64 base + VGPR32 byte offset ; LDS dest = per-lane VGPR offset
__device__ __forceinline__ void async_x16(unsigned ldsOff, const float* base, int byteOff) {
  asm volatile("global_load_async_to_lds_b128 %0, %1, %2"
               :: "v"(ldsOff), "v"(byteOff), "s"(base)
               : "memory");
}
__device__ __forceinline__ void wait_async_le1() {
  asm volatile("s_wait_asynccnt 0x1" ::: "memory");
}
__device__ __forceinline__ void wait_async_le0() {
  asm volatile("s_wait_asynccnt 0x0" ::: "memory");
}

// Evaluate spline via per-interval cubic: out = ((a3*t+a2)*t+a1)*t+a0
__device__ __forceinline__ float eval1(float xv, float invh, float c, const float4* tbl) {
  float s  = __fmaf_rn(xv, invh, c);      // (x - t0) / h
  float fl = floorf(s);
  float t  = s - fl;
  int   j  = (int)fl;
  bool  valid = ((unsigned)j) < 19u;      // 19 knot intervals; else basis == 0
  int   jc = valid ? j : 0;
  float4 a = tbl[jc];                     // ds_load_b128
  float r = __fmaf_rn(__fmaf_rn(__fmaf_rn(a.w, t, a.z), t, a.y), t, a.x);
  return valid ? r : 0.0f;
}

__device__ __forceinline__ float4 eval4(float4 xv, float invh, float c, const float4* tbl) {
  float4 r;
  r.x = eval1(xv.x, invh, c, tbl);
  r.y = eval1(xv.y, invh, c, tbl);
  r.z = eval1(xv.z, invh, c, tbl);
  r.w = eval1(xv.w, invh, c, tbl);
  return r;
}

__global__ __launch_bounds__(BLK) void bspline_kernel(
    const float* __restrict__ x, const float* __restrict__ cp,
    const float* __restrict__ knots, float* __restrict__ out,
    int n4, int nTotal)
{
  __shared__ float4 tbl[20];              // 19 used: per-interval cubic coeffs
  __shared__ float4 xbuf[2][BLK];         // double-buffered async staging

  const int tid = threadIdx.x;

  // Build the per-interval coefficient table from control points.
  // For x in [t_j, t_j+1): contributions of cp[j-3..j] (zero-padded outside [0,16))
  // with uniform cubic B-spline pieces collapse to a single cubic in t.
  if (tid < 19) {
    int j = tid;
    float c0 = (j     < 16) ? cp[j]     : 0.0f;              // piece t^3/6
    float c1 = (j - 1 >= 0 && j - 1 < 16) ? cp[j-1] : 0.0f;  // (-3t^3+3t^2+3t+1)/6
    float c2 = (j - 2 >= 0 && j - 2 < 16) ? cp[j-2] : 0.0f;  // (3t^3-6t^2+4)/6
    float c3 = (j - 3 >= 0)               ? cp[j-3] : 0.0f;  // (1-t)^3/6
    const float sixth = 1.0f / 6.0f;
    float a3 = (c0 - 3.0f*c1 + 3.0f*c2 - c3) * sixth;
    float a2 = (c1 - 2.0f*c2 + c3) * 0.5f;
    float a1 = (c1 - c3) * 0.5f;
    float a0 = (c1 + 4.0f*c2 + c3) * sixth;
    tbl[j] = make_float4(a0, a1, a2, a3);
  }
  const float t0   = knots[0];
  const float invh = 1.0f / (knots[1] - t0);
  const float cadd = -t0 * invh;
  __syncthreads();

  float4* __restrict__ out4 = (float4*)out;

  int i = blockIdx.x * BLK + tid;
  const int stride = gridDim.x * BLK;

  // Raw 32-bit LDS byte offsets for the async data mover (low 32 bits of flat addr).
  unsigned ldsA = (unsigned)(uintptr_t)&xbuf[0][tid];
  unsigned ldsB = (unsigned)(uintptr_t)&xbuf[1][tid];

  if (i < n4) {
    async_x16(ldsA, x, i * 16);           // prologue: fill buffer 0
    int buf = 0;
    int ni = i + stride;
    // Steady state: keep exactly 2 async loads in flight, zero wasted traffic.
    while (ni < n4) {
      async_x16(buf ? ldsA : ldsB, x, ni * 16);
      wait_async_le1();                   // oldest (current buffer) has landed
      const float4* bp = buf ? &xbuf[1][tid] : &xbuf[0][tid];
      out4[i] = eval4(*bp, invh, cadd, tbl);
      buf ^= 1;
      i = ni;
      ni += stride;
    }
    // Epilogue: drain the last buffer without issuing a dummy load.
    wait_async_le0();
    const float4* bp = buf ? &xbuf[1][tid] : &xbuf[0][tid];
    out4[i] = eval4(*bp, invh, cadd, tbl);
  }

  // Scalar tail (nTotal not divisible by 4) — not hit for 4096x4096 but safe.
  int rem = nTotal & 3;
  if (rem && blockIdx.x == 0 && tid < rem) {
    int idx = (nTotal & ~3) + tid;
    out[idx] = eval1(x[idx], invh, cadd, tbl);
  }
}

extern "C" void kernel_launch(void* const* d_in, const int* in_sizes, int n_in,
                              void* d_out, int out_size, void* d_ws, size_t ws_size,
                              hipStream_t stream) {
  const float* x     = (const float*)d_in[0];
  const float* cp    = (const float*)d_in[1];
  const float* knots = (const float*)d_in[2];
  float* out = (float*)d_out;

  int n  = in_sizes[0];
  int n4 = n >> 2;

  int blocks = 4096;                       // 4 float4 iterations/thread at 16M elems
  int maxBlocks = (n4 + BLK - 1) / BLK;
  if (maxBlocks < 1) maxBlocks = 1;
  if (blocks > maxBlocks) blocks = maxBlocks;

  bspline_kernel<<<blocks, BLK, 0, stream>>>(x, cp, knots, out, n4, n);
}